// MuyGPs_layer_59390807769350
// MI455X (gfx1250) — compile-verified
//
#include <hip/hip_runtime.h>
#include <hip/hip_bf16.h>

typedef float v2f __attribute__((ext_vector_type(2)));
typedef float v8f __attribute__((ext_vector_type(8)));

#define NN 64
#define FEAT 128
#define RESP 16
#define NRHS 17            // 16 targets + Kcross
#define XN_STRIDE 132      // 128 + 4 pad: bank step 4 -> conflict-free frag loads
#define KM_STRIDE 68       // 64 + 4 pad
#define RHS_STRIDE 18      // 17 + 1 pad
#define EPS_NOISE 1e-3f
#define TINY 1e-30f
#define SQRT3F 1.7320508075688772f

__global__ void zero_sigma_kernel(float* acc) {
    if (threadIdx.x < RESP) acc[threadIdx.x] = 0.0f;
}

__global__ __launch_bounds__(256) void muygps_batch_kernel(
    const float* __restrict__ x,          // (N, 128)
    const float* __restrict__ ls_ptr,     // scalar
    const float* __restrict__ targets,    // (B, 64, 16)
    const int*   __restrict__ bidx,       // (B,)
    const int*   __restrict__ nnidx,      // (B, 64)
    float* __restrict__ pred_out,         // (B, 16)
    float* __restrict__ var_scalar,       // ws: (B,)
    float* __restrict__ sigma_acc)        // ws: (16,)
{
    __shared__ float XN[NN * XN_STRIDE];
    __shared__ float KM[NN * KM_STRIDE];
    __shared__ float RHS[NN * RHS_STRIDE];
    __shared__ float XQ[FEAT];
    __shared__ float SQ[NN];
    __shared__ float KC[NN];
    __shared__ float SQQ;
    __shared__ float diag_s;

    const int b   = blockIdx.x;
    const int tid = threadIdx.x;
    const float ls = ls_ptr[0];

    // ---- gather Xn (64 x 128) and xq into LDS, float4 coalesced ----
    {
        const int row = tid >> 2;            // 0..63, 4 threads per row
        const int q   = tid & 3;
        const int g   = nnidx[b * NN + row];
        const float4* src = (const float4*)(x + (size_t)g * FEAT);
        float4* dst = (float4*)(&XN[row * XN_STRIDE]);   // row start 16B aligned (132*4)
        #pragma unroll
        for (int i = 0; i < 8; ++i) {
            const int c4 = q * 8 + i;
            dst[c4] = src[c4];
        }
        if (tid < 32) {
            const int gq = bidx[b];
            ((float4*)XQ)[tid] = ((const float4*)(x + (size_t)gq * FEAT))[tid];
        }
    }
    __syncthreads();

    // ---- row squared norms + query norm ----
    if (tid < NN) {
        float s = 0.0f;
        const float* r = &XN[tid * XN_STRIDE];
        for (int k = 0; k < FEAT; ++k) s += r[k] * r[k];
        SQ[tid] = s;
    } else if (tid == NN) {
        float s = 0.0f;
        for (int k = 0; k < FEAT; ++k) s += XQ[k] * XQ[k];
        SQQ = s;
    }
    __syncthreads();

    // ---- Kcross: matern(dist(xq, xn_i)) ----
    if (tid < NN) {
        float dot = 0.0f;
        const float* r = &XN[tid * XN_STRIDE];
        for (int k = 0; k < FEAT; ++k) dot += XQ[k] * r[k];
        float d2 = SQQ + SQ[tid] - 2.0f * dot;
        float d  = sqrtf(fmaxf(d2, TINY));
        float s  = SQRT3F * d / ls;
        KC[tid]  = (1.0f + s) * __expf(-s);
    }
    __syncthreads();

    // ---- Gram = Xn * Xn^T via f32 WMMA (16x16x4), fused Matern + nugget ----
    {
        const int wave = tid / 32;
        const int lane = tid % 32;
        const int l15  = lane & 15;
        const int hi   = lane >> 4;        // lane half selects K {0,1} vs {2,3}

        #pragma unroll
        for (int tt = 0; tt < 2; ++tt) {
            const int t  = wave * 2 + tt;  // 16 tiles, 2 per wave
            const int ti = t >> 2;
            const int tj = t & 3;
            const float* arow = &XN[(ti * 16 + l15) * XN_STRIDE + 2 * hi];
            const float* brow = &XN[(tj * 16 + l15) * XN_STRIDE + 2 * hi];
            v8f c = {};
#if __has_builtin(__builtin_amdgcn_wmma_f32_16x16x4_f32)
            for (int kk = 0; kk < FEAT; kk += 4) {
                v2f a, bv;
                a.x  = arow[kk];  a.y  = arow[kk + 1];
                bv.x = brow[kk];  bv.y = brow[kk + 1];
                c = __builtin_amdgcn_wmma_f32_16x16x4_f32(
                        false, a, false, bv, (short)0, c, false, false);
            }
            #pragma unroll
            for (int r = 0; r < 8; ++r) {
                const int M = ti * 16 + r + hi * 8;   // C layout: VGPR r -> M=r / r+8
                const int N = tj * 16 + l15;
                float d2 = SQ[M] + SQ[N] - 2.0f * c[r];
                float d  = sqrtf(fmaxf(d2, TINY));
                float s  = SQRT3F * d / ls;
                float kv = (1.0f + s) * __expf(-s);
                if (M == N) kv += EPS_NOISE;
                KM[M * KM_STRIDE + N] = kv;
            }
#else
            // fallback: scalar FMA Gram (no wmma available)
            (void)c;
            for (int e = tid; e < NN * NN; e += 256) {
                const int M = e >> 6, N = e & 63;
                float dot = 0.0f;
                for (int k = 0; k < FEAT; ++k)
                    dot += XN[M * XN_STRIDE + k] * XN[N * XN_STRIDE + k];
                float d2 = SQ[M] + SQ[N] - 2.0f * dot;
                float d  = sqrtf(fmaxf(d2, TINY));
                float s  = SQRT3F * d / ls;
                float kv = (1.0f + s) * __expf(-s);
                if (M == N) kv += EPS_NOISE;
                KM[M * KM_STRIDE + N] = kv;
            }
            break;
#endif
        }
    }
    __syncthreads();

    // ---- RHS = [targets | Kcross]  (64 x 17) ----
    for (int i = tid; i < NN * RESP; i += 256) {
        const int n = i >> 4, r = i & 15;
        RHS[n * RHS_STRIDE + r] = targets[((size_t)b * NN + n) * RESP + r];
    }
    if (tid < NN) RHS[tid * RHS_STRIDE + RESP] = KC[tid];
    __syncthreads();

    // ---- in-place Cholesky (lower), right-looking ----
    for (int k = 0; k < NN; ++k) {
        if (tid == 0) diag_s = sqrtf(KM[k * KM_STRIDE + k]);
        __syncthreads();
        const float dk = diag_s;
        if (tid == k) KM[k * KM_STRIDE + k] = dk;
        if (tid > k && tid < NN) KM[tid * KM_STRIDE + k] /= dk;
        __syncthreads();
        const int rem = NN - 1 - k;
        for (int e = tid; e < rem * rem; e += 256) {
            const int i = k + 1 + e / rem;
            const int j = k + 1 + e % rem;
            KM[i * KM_STRIDE + j] -= KM[i * KM_STRIDE + k] * KM[j * KM_STRIDE + k];
        }
        __syncthreads();
    }

    // ---- forward substitution: L * Z = RHS ----
    for (int k = 0; k < NN; ++k) {
        const float dk = KM[k * KM_STRIDE + k];
        if (tid < NRHS) RHS[k * RHS_STRIDE + tid] /= dk;
        __syncthreads();
        const int rows = NN - 1 - k;
        for (int e = tid; e < rows * NRHS; e += 256) {
            const int i  = k + 1 + e / NRHS;
            const int cc = e % NRHS;
            RHS[i * RHS_STRIDE + cc] -= KM[i * KM_STRIDE + k] * RHS[k * RHS_STRIDE + cc];
        }
        __syncthreads();
    }

    // ---- back substitution: L^T * Z = Z1 ----
    for (int k = NN - 1; k >= 0; --k) {
        const float dk = KM[k * KM_STRIDE + k];
        if (tid < NRHS) RHS[k * RHS_STRIDE + tid] /= dk;
        __syncthreads();
        for (int e = tid; e < k * NRHS; e += 256) {
            const int i  = e / NRHS;          // 0..k-1
            const int cc = e % NRHS;
            RHS[i * RHS_STRIDE + cc] -= KM[k * KM_STRIDE + i] * RHS[k * RHS_STRIDE + cc];
        }
        __syncthreads();
    }

    // ---- outputs: predictions, sigma partials, variance scalar ----
    if (tid < RESP) {
        float p = 0.0f, sg = 0.0f;
        for (int n = 0; n < NN; ++n) {
            const float z = RHS[n * RHS_STRIDE + tid];
            p  += KC[n] * z;
            sg += targets[((size_t)b * NN + n) * RESP + tid] * z;
        }
        pred_out[(size_t)b * RESP + tid] = p;
        atomicAdd(&sigma_acc[tid], sg);
    } else if (tid == RESP) {
        float v = 0.0f;
        for (int n = 0; n < NN; ++n) v += KC[n] * RHS[n * RHS_STRIDE + RESP];
        var_scalar[b] = 1.0f - v;
    }
}

__global__ void finalize_kernel(const float* __restrict__ sigma_acc,
                                const float* __restrict__ var_scalar,
                                float* __restrict__ var_out,
                                float* __restrict__ sigma_out,
                                int B)
{
    const float inv = 1.0f / ((float)NN * (float)B);
    const int i = blockIdx.x * blockDim.x + threadIdx.x;
    if (i < RESP) sigma_out[i] = sigma_acc[i] * inv;
    if (i < B * RESP) {
        const int b = i >> 4, r = i & 15;
        var_out[i] = var_scalar[b] * (sigma_acc[r] * inv);
    }
}

extern "C" void kernel_launch(void* const* d_in, const int* in_sizes, int n_in,
                              void* d_out, int out_size, void* d_ws, size_t ws_size,
                              hipStream_t stream) {
    const float* x       = (const float*)d_in[0];
    const float* ls      = (const float*)d_in[1];
    const float* targets = (const float*)d_in[2];
    const int*   bidx    = (const int*)d_in[3];
    const int*   nnidx   = (const int*)d_in[4];
    const int B = in_sizes[3];                 // 8192

    float* pred  = (float*)d_out;              // (B,16)
    float* var   = pred + (size_t)B * RESP;    // (B,16)
    float* sigma = var  + (size_t)B * RESP;    // (16,)

    float* sigma_acc  = (float*)d_ws;          // 16 floats
    float* var_scalar = sigma_acc + RESP;      // B floats

    zero_sigma_kernel<<<1, 32, 0, stream>>>(sigma_acc);
    muygps_batch_kernel<<<B, 256, 0, stream>>>(x, ls, targets, bidx, nnidx,
                                               pred, var_scalar, sigma_acc);
    const int total = B * RESP;
    finalize_kernel<<<(total + 255) / 256, 256, 0, stream>>>(sigma_acc, var_scalar,
                                                             var, sigma, B);
}